// MyModel_87522843560267
// MI455X (gfx1250) — compile-verified
//
#include <hip/hip_runtime.h>

// Bidirectional masked LSTM, H=10, F=1, B=1024, T=4096.
// One wave (32 threads) per (direction, 16-batch tile): 128 waves total.
// z^T = R^T (40x10) * h^T (10x16) via V_WMMA_F32_16X16X4_F32, batch in lanes.

typedef __attribute__((ext_vector_type(2))) float v2f;
typedef __attribute__((ext_vector_type(8))) float v8f;

#define L2E 1.44269504088896340f

__device__ __forceinline__ float fexp2(float x) { return __builtin_amdgcn_exp2f(x); }
__device__ __forceinline__ float frcp(float x)  { return __builtin_amdgcn_rcpf(x); }
__device__ __forceinline__ float sigm(float z)  { return frcp(1.0f + fexp2(-z * L2E)); }
__device__ __forceinline__ float ftanh(float z) {
    // tanh(z) = 1 - 2/(exp2(2z*log2e)+1)
    return __builtin_fmaf(-2.0f, frcp(1.0f + fexp2(z * (2.0f * L2E))), 1.0f);
}
// Swap lower/upper half-wave (lane ^ 16): ds_swizzle group-of-32, xor=0x10, and=0x1f
__device__ __forceinline__ float swap16(float v) {
    return __int_as_float(__builtin_amdgcn_ds_swizzle(__float_as_int(v), 0x401f));
}

__global__ __launch_bounds__(32) void lstm_bidir_wmma(
    const float* __restrict__ x, const int* __restrict__ mask,
    const float* __restrict__ kf, const float* __restrict__ rf, const float* __restrict__ bf,
    const float* __restrict__ kb, const float* __restrict__ rb, const float* __restrict__ bb,
    float* __restrict__ out, int B, int T)
{
    const int lane   = threadIdx.x & 31;
    const int half   = lane >> 4;     // 0 or 1
    const int bcol   = lane & 15;     // batch column within tile (N index)
    const bool lanelo = (half == 0);

    const int ntiles = B >> 4;
    const int blk  = blockIdx.x;
    const int dir  = (blk >= ntiles) ? 1 : 0;
    const int tile = dir ? (blk - ntiles) : blk;

    const float* rk = dir ? rb : rf;
    const float* kn = dir ? kb : kf;
    const float* bs = dir ? bb : bf;

    // --- Preload A = R^T per gate tile, with kernel folded in at k=10 and bias at k=11.
    // A 16x4 f32 layout: lane = row j (mod 16), VGPR r: K = 4*kk + r + 2*half.
    float A[4][3][2];
#pragma unroll
    for (int g = 0; g < 4; ++g)
#pragma unroll
        for (int kk = 0; kk < 3; ++kk)
#pragma unroll
            for (int r = 0; r < 2; ++r) {
                const int k = 4 * kk + r + 2 * half;
                const int j = 10 * g + bcol;          // gate-major column in original order
                float v = 0.0f;
                if (bcol < 10) {
                    if (k < 10)       v = rk[k * 40 + j];   // R[k, j]
                    else if (k == 10) v = kn[j];            // input kernel row (F=1)
                    else              v = bs[j];            // bias row (pairs with B=1.0)
                }
                A[g][kk][r] = v;
            }

    // --- Per-lane sequence pointers (batch b = lane%16 of this tile).
    const int b = (tile << 4) + bcol;
    const float* px = x    + (size_t)b * (size_t)T;
    const int*   pm = mask + (size_t)b * (size_t)T;

    // h, c in D-layout: VGPR v, lane -> (jj = v + 8*half, batch = bcol); jj<10 valid.
    float h[8], c[8];
#pragma unroll
    for (int v = 0; v < 8; ++v) { h[v] = 0.0f; c[v] = 0.0f; }

    const int col0 = dir ? (T - 1) : 0;
    float x_cur = px[col0];
    int   m_cur = pm[col0];

    for (int t = 0; t < T; ++t) {
        // Prefetch next step's x/mask (off the recurrent critical path).
        const int tn   = (t + 1 < T) ? (t + 1) : t;
        const int coln = dir ? (T - 1 - tn) : tn;
        const float x_nxt = px[coln];
        const int   m_nxt = pm[coln];

        // --- Build B = h^T (4x16 per K-group): relayout D->B with half-wave swaps.
        // Computed unconditionally (convergent ds_swizzle), selected per half-wave.
        const float s0 = swap16(h[0]);
        const float s1 = swap16(h[1]);
        const float s2 = swap16(h[2]);
        const float s3 = swap16(h[3]);
        const float s6 = swap16(h[6]);
        const float s7 = swap16(h[7]);
        const float B00 = lanelo ? h[0] : s2;      // k = 0 | 2
        const float B01 = lanelo ? h[1] : s3;      // k = 1 | 3
        const float B10 = lanelo ? h[4] : s6;      // k = 4 | 6
        const float B11 = lanelo ? h[5] : s7;      // k = 5 | 7
        const float B20 = lanelo ? s0 : x_cur;     // k = 8 | 10 (= x_t slot)
        const float B21 = lanelo ? s1 : 1.0f;      // k = 9 | 11 (= bias slot)
        const v2f Bk0 = { B00, B01 };
        const v2f Bk1 = { B10, B11 };
        const v2f Bk2 = { B20, B21 };

        // --- z tiles: 4 gate tiles x 3 chained K=4 WMMAs, C starts at zero.
        v8f z[4];
#pragma unroll
        for (int g = 0; g < 4; ++g) {
            v8f acc = { 0.f, 0.f, 0.f, 0.f, 0.f, 0.f, 0.f, 0.f };
            v2f a0 = { A[g][0][0], A[g][0][1] };
            v2f a1 = { A[g][1][0], A[g][1][1] };
            v2f a2 = { A[g][2][0], A[g][2][1] };
            acc = __builtin_amdgcn_wmma_f32_16x16x4_f32(false, a0, false, Bk0, (short)0, acc, false, false);
            acc = __builtin_amdgcn_wmma_f32_16x16x4_f32(false, a1, false, Bk1, (short)0, acc, false, false);
            acc = __builtin_amdgcn_wmma_f32_16x16x4_f32(false, a2, false, Bk2, (short)0, acc, false, false);
            z[g] = acc;
        }

        // --- Gates + state update (Keras order i, f, g, o), masked copy-through.
        const bool upd = (m_cur != 0);
#pragma unroll
        for (int v = 0; v < 8; ++v) {
            const float ig = sigm(z[0][v]);
            const float fg = sigm(z[1][v]);
            const float gg = ftanh(z[2][v]);
            const float og = sigm(z[3][v]);
            const float cn = __builtin_fmaf(fg, c[v], ig * gg);
            const float hn = og * ftanh(cn);
            c[v] = upd ? cn : c[v];
            h[v] = upd ? hn : h[v];
        }

        x_cur = x_nxt;
        m_cur = m_nxt;
    }

    // --- Write h: out[b, dir*10 + jj], jj = v + 8*half (valid jj < 10).
    float* po = out + (size_t)b * 20 + dir * 10;
#pragma unroll
    for (int v = 0; v < 8; ++v) {
        const int jj = v + 8 * half;
        if (jj < 10) po[jj] = h[v];
    }
}

extern "C" void kernel_launch(void* const* d_in, const int* in_sizes, int n_in,
                              void* d_out, int out_size, void* d_ws, size_t ws_size,
                              hipStream_t stream) {
    const float* x    = (const float*)d_in[0];
    const int*   mask = (const int*)d_in[1];
    const float* kf   = (const float*)d_in[2];
    const float* rf   = (const float*)d_in[3];
    const float* bf   = (const float*)d_in[4];
    const float* kb   = (const float*)d_in[5];
    const float* rb   = (const float*)d_in[6];
    const float* bb   = (const float*)d_in[7];
    float* out = (float*)d_out;

    const int T = 4096;
    const int B = in_sizes[0] / T;      // 1024
    const int ntiles = B / 16;          // 64 per direction

    dim3 grid(2 * ntiles), block(32);
    hipLaunchKernelGGL(lstm_bidir_wmma, grid, block, 0, stream,
                       x, mask, kf, rf, bf, kb, rb, bb, out, B, T);
}